// N2PAttention_68719477132
// MI455X (gfx1250) — compile-verified
//
#include <hip/hip_runtime.h>
#include <hip/hip_bf16.h>

// ---------------------------------------------------------------------------
// N2PAttention for MI455X (gfx1250, wave32, WMMA + TDM).
// - All matmuls: V_WMMA_F32_16X16X4_F32 (fp32 matrix path: KNN argmax is
//   discrete and BN stats are sensitive; total compute ~20 GFLOP, all data
//   L2-resident, so fp32 WMMA is the right precision/perf point).
// - KNN gram + top-32 fused (never materializes the 536MB neg_d tensor);
//   B-panel streamed into LDS by the Tensor Data Mover (double-buffered,
//   TENSORcnt-fenced) and consumed as ds_load_b64 fragments.
// - k/v projections use linearity: k_j = P_k[idx_j] - P_k[n].
// ---------------------------------------------------------------------------

#define B_ 8
#define C_ 64
#define N_ 4096
#define K_ 32
#define H_ 4
#define D_ 16
#define C4_ 256
#define NCHUNK 64                      // BN reduction chunks (deterministic tree)
#define BN_ROWS (B_ * N_)              // 32768 rows for BatchNorm stats
#define KNN_ROWS 64                    // TDM chunk: 64 feat rows = 16KB tile

typedef float v2f __attribute__((ext_vector_type(2)));
typedef float v8f __attribute__((ext_vector_type(8)));
typedef unsigned int v4u __attribute__((ext_vector_type(4)));
typedef int v4i __attribute__((ext_vector_type(4)));
typedef int v8i __attribute__((ext_vector_type(8)));

// ---------------------------------------------------------------------------
// TDM: async 2D tile load Global -> LDS (D# per CDNA5 ISA ch.8).
//   rows x 64 fp32 elements, row stride 64 elements, contiguous in LDS.
// ---------------------------------------------------------------------------
__device__ __forceinline__ void tdm_load_tile64(const float* gsrc,
                                                unsigned ldsByteOff,
                                                unsigned rows) {
  unsigned long long ga = (unsigned long long)(size_t)gsrc;
  v4u g0;
  g0[0] = 1u;                                        // count=1, user descriptor
  g0[1] = ldsByteOff;                                // lds_addr [63:32]
  g0[2] = (unsigned)(ga & 0xffffffffu);              // global_addr lo
  g0[3] = (unsigned)((ga >> 32) & 0x01ffffffu)       // global_addr [56:32]
        | (2u << 30);                                // type=2 ("image")
  const unsigned td0 = 64u;                          // tensor_dim0 (elements)
  const unsigned td1 = (unsigned)N_;                 // tensor_dim1
  const unsigned tile0 = 64u, tile2 = 0u;
  const unsigned stride0 = 64u;                      // dim0 stride (elements)
  v8i g1;
  g1[0] = (int)(2u << 16);                           // wg_mask=0, data_size=4B
  g1[1] = (int)((td0 & 0xffffu) << 16);              // atomic_addr=0 | td0 lo16
  g1[2] = (int)(((td0 >> 16) & 0xffffu) | ((td1 & 0xffffu) << 16));
  g1[3] = (int)(((td1 >> 16) & 0xffffu) | (tile0 << 16));
  g1[4] = (int)((rows & 0xffffu) | (tile2 << 16));   // tile_dim1 | tile_dim2
  g1[5] = (int)stride0;                              // tensor_dim0_stride lo32
  g1[6] = 0;                                         // stride0 hi16 | stride1 lo16
  g1[7] = 0;                                         // stride1 hi32
  v4i g2 = {0, 0, 0, 0};                             // dims 2/3 unused (2D)
  v4i g3 = {0, 0, 0, 0};
#if defined(__clang_major__) && (__clang_major__ >= 23)
  v8i g4 = {0, 0, 0, 0, 0, 0, 0, 0};
  __builtin_amdgcn_tensor_load_to_lds(g0, g1, g2, g3, g4, 0);
#else
  __builtin_amdgcn_tensor_load_to_lds(g0, g1, g2, g3, 0);
#endif
}

// ---------------------------------------------------------------------------
// 0) transpose x (B,C,N) -> feat (B,N,C) and per-point squared norms
// ---------------------------------------------------------------------------
__global__ void transpose_sq_kernel(const float* __restrict__ x,
                                    float* __restrict__ feat,
                                    float* __restrict__ sq) {
  int p = blockIdx.x * blockDim.x + threadIdx.x;   // 0..B*N-1
  if (p >= B_ * N_) return;
  int b = p >> 12;                                  // p / N_
  int n = p & (N_ - 1);
  const float* xb = x + (size_t)b * C_ * N_ + n;    // stride N_ over channels
  float* fr = feat + (size_t)p * C_;
  float s = 0.f;
#pragma unroll
  for (int c = 0; c < C_; ++c) {
    float v = xb[(size_t)c * N_];
    fr[c] = v;
    s += v * v;
  }
  sq[p] = s;
}

// ---------------------------------------------------------------------------
// 1) Generic GEMM  Out[m][o] = act( sum_k A[m][k] * W[o][k] )
//    One wave per 16x16 tile; V_WMMA_F32_16X16X4_F32 over K.
//    Fragment layouts per CDNA5 ISA 7.12.2:
//      A 16x4 : lanes 0-15 M=lane K={k0,k0+1}; lanes 16-31 M=lane-16 K={k0+2,k0+3}
//      B 4x16 : lanes 0-15 N=lane K={k0,k0+1}; lanes 16-31 N=lane-16 K={k0+2,k0+3}
//      D 16x16: vgpr r -> M = r + 8*(lane>=16), N = lane&15
// ---------------------------------------------------------------------------
__global__ void gemm_nt_wmma(const float* __restrict__ A,
                             const float* __restrict__ W,
                             float* __restrict__ Out,
                             int Kc, int Nc, int act) {
  const int lane = threadIdx.x;
  const int half = lane >> 4;
  const int l15  = lane & 15;
  const int m0 = blockIdx.x * 16;
  const int n0 = blockIdx.y * 16;

  const float* Arow = A + (size_t)(m0 + l15) * Kc + half * 2;
  const float* Wrow = W + (size_t)(n0 + l15) * Kc + half * 2;

  v8f acc = {};
  for (int k0 = 0; k0 < Kc; k0 += 4) {
    v2f a, b;
    a.x = Arow[k0];  a.y = Arow[k0 + 1];
    b.x = Wrow[k0];  b.y = Wrow[k0 + 1];
    acc = __builtin_amdgcn_wmma_f32_16x16x4_f32(false, a, false, b,
                                                (short)0, acc, false, false);
  }

  float* Obase = Out + (size_t)m0 * Nc + n0;
#pragma unroll
  for (int r = 0; r < 8; ++r) {
    float v = acc[r];
    if (act == 1) v = v > 0.f ? v : 0.2f * v;      // LeakyReLU(0.2)
    Obase[(size_t)(r + half * 8) * Nc + l15] = v;
  }
}

// ---------------------------------------------------------------------------
// 2) Fused KNN: WMMA gram (16 rows x all 4096 cols) + running top-32 per row.
//    neg_d(n,m) = 2*F_n.F_m - |F_n|^2 - |F_m|^2 ; keep the largest-32 index
//    set (downstream sum over K is permutation-invariant).
//    One wave per block. The 4096x64 B-panel is streamed by the TDM into a
//    double-buffered 16KB LDS tile (TENSORcnt-fenced) overlapping the 64
//    WMMAs + top-k maintenance of the previous chunk.
// ---------------------------------------------------------------------------
__global__ void knn_topk_kernel(const float* __restrict__ feat,
                                const float* __restrict__ sq,
                                int* __restrict__ idxOut) {
  __shared__ float bbuf[2][KNN_ROWS * C_];   // 2 x 16KB TDM landing buffers
  __shared__ float sc[16][17];               // score tile (padded)
  __shared__ float topv[16][K_];
  __shared__ int   topi[16][K_];

  const int lane = threadIdx.x;
  const int half = lane >> 4;
  const int l15  = lane & 15;
  const int b  = blockIdx.y;
  const int r0 = blockIdx.x * 16;

  const float* Fb  = feat + (size_t)b * N_ * C_;
  const float* sqb = sq + (size_t)b * N_;

  // A fragments: this block's 16 rows, all 64 channels (16 K-steps), in regs
  v2f areg[16];
  {
    const float* Arow = Fb + (size_t)(r0 + l15) * C_ + half * 2;
#pragma unroll
    for (int ks = 0; ks < 16; ++ks) {
      areg[ks].x = Arow[ks * 4];
      areg[ks].y = Arow[ks * 4 + 1];
    }
  }
  const float myrowsq = sqb[r0 + l15];   // meaningful for owner lanes (<16)

  if (lane < 16) {
#pragma unroll
    for (int j = 0; j < K_; ++j) { topv[lane][j] = -3.4e38f; topi[lane][j] = j; }
  }
  float curmin = -3.4e38f;
  int   curpos = 0;
  __syncthreads();

  // prime the pipeline: DMA chunk 0 into buffer 0
  tdm_load_tile64(Fb, (unsigned)(size_t)&bbuf[0][0], KNN_ROWS);

  const int nch = N_ / KNN_ROWS;                    // 64 chunks
  for (int ch = 0; ch < nch; ++ch) {
    const int cur = ch & 1;
    __builtin_amdgcn_s_wait_tensorcnt(0);           // chunk ch landed
    asm volatile("" ::: "memory");                  // TDM wrote LDS: no folding
    if (ch + 1 < nch)                               // overlap DMA of ch+1
      tdm_load_tile64(Fb + (size_t)(ch + 1) * KNN_ROWS * C_,
                      (unsigned)(size_t)&bbuf[cur ^ 1][0], KNN_ROWS);

    const float* Bl = &bbuf[cur][0];
#pragma unroll
    for (int t = 0; t < 4; ++t) {                   // four 16-col sub-tiles
      const int c0 = ch * KNN_ROWS + t * 16;
      const float* Brow = Bl + (t * 16 + l15) * C_ + half * 2;  // LDS reads

      v8f acc = {};
#pragma unroll
      for (int ks = 0; ks < 16; ++ks) {
        v2f bf;
        bf.x = Brow[ks * 4];
        bf.y = Brow[ks * 4 + 1];
        acc = __builtin_amdgcn_wmma_f32_16x16x4_f32(false, areg[ks], false, bf,
                                                    (short)0, acc, false, false);
      }

      const float colsq = sqb[c0 + l15];
#pragma unroll
      for (int r = 0; r < 8; ++r)
        sc[r + half * 8][l15] = 2.f * acc[r] - colsq;  // row term by owner
      __syncthreads();

      if (lane < 16) {
        const int m = lane;
#pragma unroll 4
        for (int j = 0; j < 16; ++j) {
          float v = sc[m][j] - myrowsq;
          if (v > curmin) {
            topv[m][curpos] = v;
            topi[m][curpos] = c0 + j;
            float mn = topv[m][0]; int mp = 0;
#pragma unroll
            for (int tt = 1; tt < K_; ++tt) {
              float tv = topv[m][tt];
              if (tv < mn) { mn = tv; mp = tt; }
            }
            curmin = mn; curpos = mp;
          }
        }
      }
      __syncthreads();
    }
  }

  if (lane < 16) {
    int* dst = idxOut + ((size_t)b * N_ + r0 + lane) * K_;
#pragma unroll
    for (int j = 0; j < K_; ++j) dst[j] = topi[lane][j];
  }
}

// ---------------------------------------------------------------------------
// 3) Attention: one wave per point, lane j <-> neighbor j (K=32=warpSize).
//    k_j = Pk[idx_j]-Pk[n], v_j = Pv[idx_j]-Pv[n]; shfl softmax per head.
// ---------------------------------------------------------------------------
__global__ void attn_kernel(const float* __restrict__ q,
                            const float* __restrict__ pk,
                            const float* __restrict__ pv,
                            const int* __restrict__ idx,
                            float* __restrict__ outattn) {
  __shared__ float qs[4][C_], pkn[4][C_], pvn[4][C_];
  const int wid  = threadIdx.x >> 5;
  const int lane = threadIdx.x & 31;
  const int point = blockIdx.x * 4 + wid;          // 0..B*N-1
  const int b = point >> 12;

  qs [wid][lane] = q [(size_t)point * C_ + lane];
  qs [wid][lane + 32] = q [(size_t)point * C_ + lane + 32];
  pkn[wid][lane] = pk[(size_t)point * C_ + lane];
  pkn[wid][lane + 32] = pk[(size_t)point * C_ + lane + 32];
  pvn[wid][lane] = pv[(size_t)point * C_ + lane];
  pvn[wid][lane + 32] = pv[(size_t)point * C_ + lane + 32];
  __syncthreads();

  const int jn = idx[(size_t)point * K_ + lane];           // neighbor (in-batch)
  const size_t grow = ((size_t)b * N_ + jn) * C_;
  const float* pkj = pk + grow;
  const float* pvj = pv + grow;
  __builtin_prefetch(pvj, 0, 0);                   // global_prefetch_b8

  float e0 = 0.f, e1 = 0.f, e2 = 0.f, e3 = 0.f;
#pragma unroll
  for (int c = 0; c < C_; ++c) {
    float kd = pkj[c] - pkn[wid][c];
    float t = qs[wid][c] * kd;
    if (c < 16) e0 += t; else if (c < 32) e1 += t; else if (c < 48) e2 += t; else e3 += t;
  }
  e0 *= 0.25f; e1 *= 0.25f; e2 *= 0.25f; e3 *= 0.25f;     // 1/sqrt(D)

  float a0, a1, a2, a3;
  {
    float m0 = e0, m1 = e1, m2 = e2, m3 = e3;
#pragma unroll
    for (int off = 16; off > 0; off >>= 1) {
      m0 = fmaxf(m0, __shfl_xor(m0, off, 32));
      m1 = fmaxf(m1, __shfl_xor(m1, off, 32));
      m2 = fmaxf(m2, __shfl_xor(m2, off, 32));
      m3 = fmaxf(m3, __shfl_xor(m3, off, 32));
    }
    a0 = __expf(e0 - m0); a1 = __expf(e1 - m1);
    a2 = __expf(e2 - m2); a3 = __expf(e3 - m3);
    float s0 = a0, s1 = a1, s2 = a2, s3 = a3;
#pragma unroll
    for (int off = 16; off > 0; off >>= 1) {
      s0 += __shfl_xor(s0, off, 32);
      s1 += __shfl_xor(s1, off, 32);
      s2 += __shfl_xor(s2, off, 32);
      s3 += __shfl_xor(s3, off, 32);
    }
    a0 /= s0; a1 /= s1; a2 /= s2; a3 /= s3;
  }

  float* orow = outattn + (size_t)point * C_;
#pragma unroll
  for (int c = 0; c < C_; ++c) {
    float vd = pvj[c] - pvn[wid][c];
    float ah = (c < 16) ? a0 : (c < 32) ? a1 : (c < 48) ? a2 : a3;
    float contrib = ah * vd;
#pragma unroll
    for (int off = 16; off > 0; off >>= 1)
      contrib += __shfl_xor(contrib, off, 32);
    if (lane == 0) orow[c] = contrib;
  }
}

// ---------------------------------------------------------------------------
// 4) BatchNorm: deterministic two-stage reduction over (B*N) per channel.
// ---------------------------------------------------------------------------
__global__ void bn_stats_partial(const float* __restrict__ A,
                                 const float* __restrict__ Bx,
                                 float2* __restrict__ partial) {
  __shared__ float sh[4][C_], sh2[4][C_];
  const int c  = threadIdx.x & 63;
  const int rg = threadIdx.x >> 6;                  // 0..3
  const int rows = BN_ROWS / NCHUNK;                // 512
  const int r0 = blockIdx.x * rows;
  float s = 0.f, s2 = 0.f;
  for (int r = rg; r < rows; r += 4) {
    size_t i = (size_t)(r0 + r) * C_ + c;
    float v = A[i] + Bx[i];
    s += v; s2 += v * v;
  }
  sh[rg][c] = s; sh2[rg][c] = s2;
  __syncthreads();
  if (rg == 0) {
    s  = sh [0][c] + sh [1][c] + sh [2][c] + sh [3][c];
    s2 = sh2[0][c] + sh2[1][c] + sh2[2][c] + sh2[3][c];
    partial[blockIdx.x * C_ + c] = make_float2(s, s2);
  }
}

__global__ void bn_finalize(const float2* __restrict__ partial,
                            const float* __restrict__ g,
                            const float* __restrict__ bb,
                            float* __restrict__ scale,
                            float* __restrict__ shift) {
  const int c = threadIdx.x;                        // 64 threads
  float s = 0.f, s2 = 0.f;
  for (int i = 0; i < NCHUNK; ++i) {                // fixed order: deterministic
    float2 p = partial[i * C_ + c];
    s += p.x; s2 += p.y;
  }
  const float inv_n = 1.f / (float)BN_ROWS;
  float mu  = s * inv_n;
  float var = s2 * inv_n - mu * mu;                 // biased, torch-style
  float iv  = rsqrtf(var + 1e-5f);
  float sc  = g[c] * iv;
  scale[c] = sc;
  shift[c] = bb[c] - mu * sc;
}

__global__ void bn_apply(const float* __restrict__ A, const float* __restrict__ Bx,
                         const float* __restrict__ scale, const float* __restrict__ shift,
                         float* __restrict__ Out) {
  size_t i = (size_t)blockIdx.x * blockDim.x + threadIdx.x;
  int c = i & 63;
  Out[i] = (A[i] + Bx[i]) * scale[c] + shift[c];
}

// Final BN + residual + transpose back to (B, C, N), straight into d_out.
__global__ void bn_apply_transpose(const float* __restrict__ A, const float* __restrict__ Bx,
                                   const float* __restrict__ scale, const float* __restrict__ shift,
                                   float* __restrict__ Out) {
  size_t i = (size_t)blockIdx.x * blockDim.x + threadIdx.x;  // index in (b,c,n)
  int n = i & (N_ - 1);
  int c = (i >> 12) & 63;
  int b = (int)(i >> 18);
  size_t src = ((size_t)b * N_ + n) * C_ + c;
  Out[i] = (A[src] + Bx[src]) * scale[c] + shift[c];
}

// ---------------------------------------------------------------------------
extern "C" void kernel_launch(void* const* d_in, const int* in_sizes, int n_in,
                              void* d_out, int out_size, void* d_ws, size_t ws_size,
                              hipStream_t stream) {
  const float* x  = (const float*)d_in[0];
  const float* wq = (const float*)d_in[1];
  const float* wk = (const float*)d_in[2];
  const float* wv = (const float*)d_in[3];
  const float* w1 = (const float*)d_in[4];
  const float* w2 = (const float*)d_in[5];
  const float* g1 = (const float*)d_in[6];
  const float* b1 = (const float*)d_in[7];
  const float* g2 = (const float*)d_in[8];
  const float* b2 = (const float*)d_in[9];

  const size_t BNC = (size_t)B_ * N_ * C_;          // 2,097,152 floats
  float* ws   = (float*)d_ws;                       // ~64 MB used
  float* feat = ws;                                 // (B,N,C)
  float* qb   = ws + 1 * BNC;                       // q projection
  float* pk   = ws + 2 * BNC;                       // feat @ Wk^T
  float* pv   = ws + 3 * BNC;                       // feat @ Wv^T
  float* att  = ws + 4 * BNC;                       // attention output
  float* f1   = ws + 5 * BNC;                       // BN1 result
  float* ffb  = ws + 6 * BNC;                       // MLP output
  float* hid  = ws + 1 * BNC;                       // reuse q..att (= B*N*256)
  float* sq   = ws + 7 * BNC;                       // |f|^2  (B*N)
  int*   idx  = (int*)(ws + 7 * BNC + BN_ROWS);     // (B,N,K)
  float* part = ws + 7 * BNC + BN_ROWS + (size_t)B_ * N_ * K_;  // float2-aligned
  float* sc1  = part + 2 * NCHUNK * C_;
  float* sh1  = sc1 + C_;
  float* sc2  = sh1 + C_;
  float* sh2  = sc2 + C_;

  // 0) layout + norms
  transpose_sq_kernel<<<(B_ * N_ + 255) / 256, 256, 0, stream>>>(x, feat, sq);

  // 1) single-point projections (linearity trick replaces per-neighbor GEMMs)
  dim3 g64(BN_ROWS / 16, C_ / 16);
  gemm_nt_wmma<<<g64, 32, 0, stream>>>(feat, wq, qb, C_, C_, 0);
  gemm_nt_wmma<<<g64, 32, 0, stream>>>(feat, wk, pk, C_, C_, 0);
  gemm_nt_wmma<<<g64, 32, 0, stream>>>(feat, wv, pv, C_, C_, 0);

  // 2) fused gram + top-32 (TDM-streamed B panel, never materializes neg_d)
  knn_topk_kernel<<<dim3(N_ / 16, B_), 32, 0, stream>>>(feat, sq, idx);

  // 3) attention (one wave per point)
  attn_kernel<<<BN_ROWS / 4, 128, 0, stream>>>(qb, pk, pv, idx, att);

  // 4) BN1 over feat + att
  bn_stats_partial<<<NCHUNK, 256, 0, stream>>>(feat, att, (float2*)part);
  bn_finalize<<<1, C_, 0, stream>>>((float2*)part, g1, b1, sc1, sh1);
  bn_apply<<<(int)(BNC / 256), 256, 0, stream>>>(feat, att, sc1, sh1, f1);

  // 5) MLP: 64 -> 256 (LeakyReLU fused) -> 64
  gemm_nt_wmma<<<dim3(BN_ROWS / 16, C4_ / 16), 32, 0, stream>>>(f1, w1, hid, C_, C4_, 1);
  gemm_nt_wmma<<<g64, 32, 0, stream>>>(hid, w2, ffb, C4_, C_, 0);

  // 6) BN2 over f1 + ff, fused with the final transpose to (B,C,N)
  bn_stats_partial<<<NCHUNK, 256, 0, stream>>>(f1, ffb, (float2*)part);
  bn_finalize<<<1, C_, 0, stream>>>((float2*)part, g2, b2, sc2, sh2);
  bn_apply_transpose<<<(int)(BNC / 256), 256, 0, stream>>>(f1, ffb, sc2, sh2, (float*)d_out);
}